// LinearDHT_45329084842268
// MI455X (gfx1250) — compile-verified
//
#include <hip/hip_runtime.h>
#include <hip/hip_bf16.h>

#define NUM_R 725
#define NUM_T 180
#define IMG_N 512

typedef __attribute__((ext_vector_type(2))) float v2f;
typedef __attribute__((ext_vector_type(8))) float v8f;

// ---------------------------------------------------------------------------
// Kernel 1: 512x512 fp32 transpose into workspace (LDS tiled, conflict-free).
// Lets the y-sweep (|cos|>|sin|) gather become lane-contiguous like the x-sweep.
// ---------------------------------------------------------------------------
__global__ __launch_bounds__(256) void hough_transpose512(
    const float* __restrict__ in, float* __restrict__ out) {
  __shared__ float tile[32][33];
  const int bx = blockIdx.x * 32;
  const int by = blockIdx.y * 32;
  const int tx = threadIdx.x;   // 0..31
  const int ty = threadIdx.y;   // 0..7
#pragma unroll
  for (int j = 0; j < 32; j += 8)
    tile[ty + j][tx] = in[(by + ty + j) * IMG_N + (bx + tx)];
  __syncthreads();
#pragma unroll
  for (int j = 0; j < 32; j += 8)
    out[(bx + ty + j) * IMG_N + (by + tx)] = tile[tx][ty + j];
}

// Gather one line sample; matches reference rounding/masking semantics.
__device__ __forceinline__ float hough_sample(const float* __restrict__ base,
                                              float rho, float mult, float den,
                                              int n) {
  const float nf = (float)n;
  float prod = nf * mult;
  // Block fma-contraction: reference rounds the mul and the sub separately.
  asm volatile("" : "+v"(prod));
  const float q = (rho - prod) / den;        // IEEE div, matches jnp.divide
  const float fy = rintf(q);                 // RNE == jnp.round
  float v = 0.0f;
  if (fy >= 0.0f && fy <= 511.0f) {
    v = base[(n << 9) + (int)fy];
  }
  return v;
}

// ---------------------------------------------------------------------------
// Kernel 2: one wave per (theta, 16-rho block). Lanes gather the WMMA A-matrix
// (16x4 f32): lanes 0-15 hold K=0,1 rows M=0..15; lanes 16-31 hold K=2,3.
// B = all-ones 4x16, so C[m][*] accumulates the row-sum over all 512 samples.
// Two independent C accumulators (even/odd n-chunks) break the WMMA->WMMA
// RAW chain so XDL latency overlaps with the div/gather VALU work.
// ---------------------------------------------------------------------------
__global__ __launch_bounds__(256) void hough_wmma_kernel(
    const float* __restrict__ img,    // row-major img[x*512 + y]
    const float* __restrict__ imgT,   // transposed: imgT[y*512 + x]
    float* __restrict__ out) {
  const int lane = threadIdx.x & 31;
  const int wid  = threadIdx.x >> 5;
  const int rblk = blockIdx.x * 8 + wid;           // 46 blocks of 16 rhos
  const int t    = blockIdx.y;                     // theta index
  if (rblk >= 46) return;                          // wave-uniform exit

  const int rbase = rblk * 16;
  const int m  = lane & 15;                        // rho row within tile
  const int kh = lane >> 4;                        // K half: 0 -> K0,1 ; 1 -> K2,3

  // theta_t = t * f32(pi/180); cos/sin in f32
  const float theta = (float)t * 0.01745329238474369f;  // f32(pi/180)
  const float s = sinf(theta);
  const float c = cosf(theta);
  const bool use_x = fabsf(s) >= fabsf(c);

  // rhos = linspace(-diag, diag, 725): step = 2*diag/724, rho = -diag + r*step
  const float diag = sqrtf(524288.0f);
  const float step = (diag + diag) / 724.0f;
  float rmul = (float)(rbase + m) * step;
  asm volatile("" : "+v"(rmul));                   // keep mul+add unfused
  const float rho = rmul - diag;

  const float mult = use_x ? c : s;
  float den        = use_x ? s : c;
  if (fabsf(den) < 1e-6f) den = 1.0f;
  const float* __restrict__ base = use_x ? img : imgT;

  v8f acc0 = {};                                   // C accumulators = 0
  v8f acc1 = {};
  const v2f ones = {1.0f, 1.0f};                   // B all-ones (layout-proof)

  for (int n0 = 0; n0 < IMG_N; n0 += 8) {
    const int na = n0 + 2 * kh;
    v2f a0, a1;
    a0.x = hough_sample(base, rho, mult, den, na);
    a0.y = hough_sample(base, rho, mult, den, na + 1);
    a1.x = hough_sample(base, rho, mult, den, na + 4);
    a1.y = hough_sample(base, rho, mult, den, na + 5);
    acc0 = __builtin_amdgcn_wmma_f32_16x16x4_f32(
        false, a0, false, ones, (short)0, acc0, false, false);
    acc1 = __builtin_amdgcn_wmma_f32_16x16x4_f32(
        false, a1, false, ones, (short)0, acc1, false, false);
  }

  const v8f acc = acc0 + acc1;

  // C layout (16x16 f32): VGPR i -> M=i (lanes 0-15), M=8+i (lanes 16-31).
  // Lane 0 writes rows 0..7, lane 16 writes rows 8..15.
  if (m == 0) {
#pragma unroll
    for (int i = 0; i < 8; ++i) {
      const int r = rbase + 8 * kh + i;
      if (r < NUM_R) out[r * NUM_T + t] = acc[i];
    }
  }
}

extern "C" void kernel_launch(void* const* d_in, const int* in_sizes, int n_in,
                              void* d_out, int out_size, void* d_ws, size_t ws_size,
                              hipStream_t stream) {
  const float* img = (const float*)d_in[0];
  float* imgT = (float*)d_ws;            // 512*512*4 = 1 MB scratch
  float* out  = (float*)d_out;           // 725*180 floats

  dim3 tb(32, 8), tg(16, 16);
  hough_transpose512<<<tg, tb, 0, stream>>>(img, imgT);

  dim3 hb(256), hg(6, NUM_T);            // 6*8 waves covers 46 rho blocks
  hough_wmma_kernel<<<hg, hb, 0, stream>>>(img, imgT, out);
}